// HeteroGAT_1477468750187
// MI455X (gfx1250) — compile-verified
//
#include <hip/hip_runtime.h>

// ---------------- problem constants ----------------
#define N_FILE   50000L
#define N_CLASS  100000L
#define N_FUNC   300000L
#define N_REL    13
#define N_EDGE   200000
#define C_HID    128          // per-head channels
#define N_HEAD   2
#define OUT_W    (N_HEAD * C_HID)   // 256 = H*HID columns of hs

static const int SRCT[N_REL] = {0,0,0,0,0,0,1,1,2,2,1,1,2};
static const int DSTT[N_REL] = {1,2,2,0,1,2,1,2,1,2,2,1,2};

typedef __attribute__((ext_vector_type(2))) float v2f;
typedef __attribute__((ext_vector_type(8))) float v8f;

// ---------------- helpers ----------------
__device__ __forceinline__ unsigned f2key(float f) {
  unsigned u = __float_as_uint(f);
  return (u & 0x80000000u) ? ~u : (u | 0x80000000u);
}
__device__ __forceinline__ float key2f(unsigned k) {
  return __uint_as_float((k & 0x80000000u) ? (k ^ 0x80000000u) : ~k);
}
__device__ __forceinline__ float lrelu(float a) { return a > 0.f ? a : 0.2f * a; }

// ---------------- generic fill ----------------
__global__ void fill_u32(unsigned* __restrict__ p, unsigned v, long n) {
  long i = (long)blockIdx.x * blockDim.x + threadIdx.x;
  if (i < n) p[i] = v;
}

// ---------------- fold: u[k][h] = sum_c W[k][h*C+c] * att[h][c] ----------------
__global__ void fold_att(const float* __restrict__ w, const float* __restrict__ att,
                         float* __restrict__ u, int K) {
  int i = blockIdx.x * blockDim.x + threadIdx.x;
  if (i >= K * N_HEAD) return;
  int k = i >> 1, h = i & 1;
  const float* wc = w + (long)k * OUT_W + h * C_HID;
  const float* ac = att + h * C_HID;
  float s = 0.f;
  for (int c = 0; c < C_HID; ++c) s += wc[c] * ac[c];
  u[k * 2 + h] = s;
}

// ---------------- GEMV: e[n][h] = X[n][:] . u[:][h]  (wave per node) ----------------
__global__ void gemv_att(const float* __restrict__ X, const float* __restrict__ u,
                         float* __restrict__ e, long N, int K) {
  __shared__ float su[512];               // K*2 <= 512
  for (int i = threadIdx.x; i < K * 2; i += blockDim.x) su[i] = u[i];
  __syncthreads();
  int wave = threadIdx.x >> 5, lane = threadIdx.x & 31;
  long node = (long)blockIdx.x * (blockDim.x >> 5) + wave;
  if (node >= N) return;
  const float* xr = X + node * K;
  float a0 = 0.f, a1 = 0.f;
  for (int k = lane; k < K; k += 32) {
    float xv = xr[k];
    a0 += xv * su[k * 2 + 0];
    a1 += xv * su[k * 2 + 1];
  }
  for (int m = 16; m >= 1; m >>= 1) {
    a0 += __shfl_xor(a0, m, 32);
    a1 += __shfl_xor(a1, m, 32);
  }
  if (lane == 0) { e[node * 2 + 0] = a0; e[node * 2 + 1] = a1; }
}

// ---------------- WMMA GEMM: Y[N][256] = X[N][K] * W[K][256] (fp32) ----------------
#define BM 128
#define BK 32
__global__ __launch_bounds__(256) void gemm_hs(const float* __restrict__ X,
                                               const float* __restrict__ W,
                                               float* __restrict__ Y, long N, int K) {
  __shared__ float Xs[BM][BK + 1];     // +1 pad: avoid bank conflicts on column reads
  __shared__ float Ws[BK][OUT_W];
  const int tid  = threadIdx.x;
  const int wave = tid >> 5, lane = tid & 31;
  const int lm = lane & 15, half = lane >> 4;
  const long m0 = (long)blockIdx.x * BM;
  const int n0w = wave * 32;           // each wave owns 32 output columns

  v8f acc[8][2];
  for (int i = 0; i < 8; ++i)
    for (int j = 0; j < 2; ++j) acc[i][j] = (v8f){0.f,0.f,0.f,0.f,0.f,0.f,0.f,0.f};

  for (int k0 = 0; k0 < K; k0 += BK) {
    // stage X tile [BM][BK]
    #pragma unroll
    for (int i = 0; i < (BM * BK) / 256; ++i) {
      int idx = tid + i * 256;
      int r = idx >> 5, c = idx & 31;
      long gr = m0 + r;
      Xs[r][c] = (gr < N) ? X[gr * K + k0 + c] : 0.f;
    }
    // stage W tile [BK][256]
    #pragma unroll
    for (int i = 0; i < (BK * OUT_W) / 256; ++i) {
      int idx = tid + i * 256;
      int r = idx >> 8, c = idx & 255;
      Ws[r][c] = W[(long)(k0 + r) * OUT_W + c];
    }
    __syncthreads();

    // warm L2/WGP$ with next chunk while this chunk computes (global_prefetch_b8)
    if (k0 + BK < K) {
      {
        int idx = tid;                      // one line per thread is plenty
        int r = idx >> 5, c = idx & 31;
        long gr = m0 + r;
        if (gr < N) __builtin_prefetch(&X[gr * K + k0 + BK + c], 0, 1);
      }
      {
        int r = tid >> 3, c = (tid & 7) * 32;
        __builtin_prefetch(&W[(long)(k0 + BK + r) * OUT_W + c], 0, 1);
      }
    }

    #pragma unroll
    for (int ks = 0; ks < BK; ks += 4) {
      // ---- batch all fragment loads first, then dense WMMA burst ----
      // B frags (K x N = 4x16): VGPR0 lanes0-15 K=ks+0, lanes16-31 K=ks+2; VGPR1 K=ks+1/ks+3
      v2f b0, b1, a[8];
      b0.x = Ws[ks + half * 2 + 0][n0w + lm];
      b0.y = Ws[ks + half * 2 + 1][n0w + lm];
      b1.x = Ws[ks + half * 2 + 0][n0w + 16 + lm];
      b1.y = Ws[ks + half * 2 + 1][n0w + 16 + lm];
      #pragma unroll
      for (int mt = 0; mt < 8; ++mt) {
        // A frag (M x K = 16x4): lanes0-15 K=ks+0/+1, lanes16-31 K=ks+2/+3
        a[mt].x = Xs[mt * 16 + lm][ks + half * 2 + 0];
        a[mt].y = Xs[mt * 16 + lm][ks + half * 2 + 1];
      }
      #pragma unroll
      for (int mt = 0; mt < 8; ++mt) {
        acc[mt][0] = __builtin_amdgcn_wmma_f32_16x16x4_f32(
            false, a[mt], false, b0, (short)0, acc[mt][0], false, false);
        acc[mt][1] = __builtin_amdgcn_wmma_f32_16x16x4_f32(
            false, a[mt], false, b1, (short)0, acc[mt][1], false, false);
      }
    }
    __syncthreads();
  }

  // store: C/D layout — VGPR i: lanes0-15 M=i, lanes16-31 M=i+8; N=lane
  #pragma unroll
  for (int mt = 0; mt < 8; ++mt)
    #pragma unroll
    for (int nt = 0; nt < 2; ++nt) {
      int col = n0w + nt * 16 + lm;
      #pragma unroll
      for (int i = 0; i < 8; ++i) {
        long row = m0 + mt * 16 + half * 8 + i;
        if (row < N) Y[row * OUT_W + col] = acc[mt][nt][i];
      }
    }
}

// ---------------- edge pass 1: segment max (ordered-uint atomicMax) ----------------
__global__ void edge_max(const float* __restrict__ es, const float* __restrict__ ed,
                         const int* __restrict__ se, const int* __restrict__ de,
                         unsigned* __restrict__ mkey, int E) {
  int i = blockIdx.x * blockDim.x + threadIdx.x;
  if (i >= E) return;
  int s = se[i], d = de[i];
  #pragma unroll
  for (int h = 0; h < N_HEAD; ++h) {
    float a = lrelu(es[s * 2 + h] + ed[d * 2 + h]);
    atomicMax(&mkey[d * 2 + h], f2key(a));
  }
}

// ---------------- edge pass 2: denominator ----------------
__global__ void edge_expsum(const float* __restrict__ es, const float* __restrict__ ed,
                            const int* __restrict__ se, const int* __restrict__ de,
                            const unsigned* __restrict__ mkey, float* __restrict__ den, int E) {
  int i = blockIdx.x * blockDim.x + threadIdx.x;
  if (i >= E) return;
  int s = se[i], d = de[i];
  #pragma unroll
  for (int h = 0; h < N_HEAD; ++h) {
    float a = lrelu(es[s * 2 + h] + ed[d * 2 + h]);
    atomicAdd(&den[d * 2 + h], __expf(a - key2f(mkey[d * 2 + h])));
  }
}

// ---------------- edge pass 3: weighted aggregation (wave per edge) ----------------
__global__ void edge_aggr(const float* __restrict__ es, const float* __restrict__ ed,
                          const int* __restrict__ se, const int* __restrict__ de,
                          const unsigned* __restrict__ mkey, const float* __restrict__ den,
                          const float* __restrict__ hs, float* __restrict__ out, int E) {
  long gid = (long)blockIdx.x * blockDim.x + threadIdx.x;
  int e = (int)(gid >> 5), lane = (int)(gid & 31);
  if (e >= E) return;
  int s = se[e], d = de[e];
  float w0, w1;
  {
    float a0 = lrelu(es[s * 2 + 0] + ed[d * 2 + 0]);
    float a1 = lrelu(es[s * 2 + 1] + ed[d * 2 + 1]);
    // 0.5f = mean over the 2 heads
    w0 = __expf(a0 - key2f(mkey[d * 2 + 0])) / den[d * 2 + 0] * 0.5f;
    w1 = __expf(a1 - key2f(mkey[d * 2 + 1])) / den[d * 2 + 1] * 0.5f;
  }
  const float* hr = hs + (long)s * OUT_W;
  float* orow = out + (long)d * C_HID;
  #pragma unroll
  for (int j = 0; j < 4; ++j) {
    int c = lane + j * 32;
    atomicAdd(&orow[c], w0 * hr[c] + w1 * hr[C_HID + c]);
  }
}

// ---------------- epilogue: + sum of biases for this dst type, ReLU ----------------
__global__ void bias_relu(float* __restrict__ y, const float* __restrict__ bias,
                          int type, long N) {
  const int DT[N_REL] = {1,2,2,0,1,2,1,2,1,2,2,1,2};
  long i = (long)blockIdx.x * blockDim.x + threadIdx.x;
  if (i >= N * C_HID) return;
  int c = (int)(i & (C_HID - 1));
  float s = y[i];
  #pragma unroll
  for (int r = 0; r < N_REL; ++r)
    if (DT[r] == type) s += bias[r * C_HID + c];
  y[i] = s > 0.f ? s : 0.f;
}

// ---------------- driver ----------------
static inline unsigned cdiv(long a, long b) { return (unsigned)((a + b - 1) / b); }

extern "C" void kernel_launch(void* const* d_in, const int* in_sizes, int n_in,
                              void* d_out, int out_size, void* d_ws, size_t ws_size,
                              hipStream_t stream) {
  (void)in_sizes; (void)n_in; (void)out_size; (void)ws_size;

  const float* x0[3]   = {(const float*)d_in[0], (const float*)d_in[1], (const float*)d_in[2]};
  const float* wsrc[2] = {(const float*)d_in[3], (const float*)d_in[8]};
  const float* wdst[2] = {(const float*)d_in[4], (const float*)d_in[9]};
  const float* asrc[2] = {(const float*)d_in[5], (const float*)d_in[10]};
  const float* adst[2] = {(const float*)d_in[6], (const float*)d_in[11]};
  const float* bias[2] = {(const float*)d_in[7], (const float*)d_in[12]};
  const int* esrc = (const int*)d_in[13];
  const int* edst = (const int*)d_in[14];

  const long NTl[3] = {N_FILE, N_CLASS, N_FUNC};

  // ---- workspace carve (floats) ----
  float* p = (float*)d_ws;
  float* x1[3];
  x1[0] = p; p += N_FILE  * C_HID;
  x1[1] = p; p += N_CLASS * C_HID;
  x1[2] = p; p += N_FUNC  * C_HID;
  float* hs = p;  p += N_FUNC * (long)OUT_W;     // largest Ns * 256
  float* es = p;  p += N_FUNC * 2;
  float* ev = p;  p += N_FUNC * 2;
  unsigned* mk = (unsigned*)p; p += N_FUNC * 2;
  float* den = p; p += N_FUNC * 2;
  float* usr = p; p += 512;
  float* uds = p; p += 512;

  float* y2[3];
  y2[0] = (float*)d_out;
  y2[1] = y2[0] + N_FILE  * C_HID;
  y2[2] = y2[1] + N_CLASS * C_HID;

  for (int L = 0; L < 2; ++L) {
    const int K = (L == 0) ? 256 : 128;
    const float* Xin[3]; float* Y[3];
    for (int t = 0; t < 3; ++t) {
      Xin[t] = (L == 0) ? x0[t] : x1[t];
      Y[t]   = (L == 0) ? x1[t] : y2[t];
    }
    // zero the accumulators for this layer's outputs
    for (int t = 0; t < 3; ++t) {
      long n = NTl[t] * C_HID;
      fill_u32<<<cdiv(n, 256), 256, 0, stream>>>((unsigned*)Y[t], 0u, n);
    }

    for (int r = 0; r < N_REL; ++r) {
      const int s = SRCT[r], d = DSTT[r];
      const long Ns = NTl[s], Nd = NTl[d];
      const float* Wr  = wsrc[L] + (long)r * K * OUT_W;
      const float* Wdr = wdst[L] + (long)r * K * OUT_W;

      // fold att into K x 2 vectors, then GEMV es / ed
      fold_att<<<cdiv(K * 2, 256), 256, 0, stream>>>(Wr,  asrc[L] + (long)r * OUT_W, usr, K);
      fold_att<<<cdiv(K * 2, 256), 256, 0, stream>>>(Wdr, adst[L] + (long)r * OUT_W, uds, K);
      gemv_att<<<cdiv(Ns, 8), 256, 0, stream>>>(Xin[s], usr, es, Ns, K);
      gemv_att<<<cdiv(Nd, 8), 256, 0, stream>>>(Xin[d], uds, ev, Nd, K);

      // hs = X_src @ W_src  (WMMA fp32)
      gemm_hs<<<cdiv(Ns, BM), 256, 0, stream>>>(Xin[s], Wr, hs, Ns, K);

      // segment softmax + aggregation
      fill_u32<<<cdiv(Nd * 2, 256), 256, 0, stream>>>(mk, 0u, Nd * 2);
      fill_u32<<<cdiv(Nd * 2, 256), 256, 0, stream>>>((unsigned*)den, 0u, Nd * 2);
      const int* se = esrc + (long)r * N_EDGE;
      const int* de = edst + (long)r * N_EDGE;
      edge_max   <<<cdiv(N_EDGE, 256), 256, 0, stream>>>(es, ev, se, de, mk, N_EDGE);
      edge_expsum<<<cdiv(N_EDGE, 256), 256, 0, stream>>>(es, ev, se, de, mk, den, N_EDGE);
      edge_aggr  <<<cdiv((long)N_EDGE * 32, 256), 256, 0, stream>>>(es, ev, se, de, mk, den,
                                                                    hs, Y[d], N_EDGE);
    }

    for (int t = 0; t < 3; ++t) {
      long n = NTl[t] * C_HID;
      bias_relu<<<cdiv(n, 256), 256, 0, stream>>>(Y[t], bias[L], t, NTl[t]);
    }
  }
}